// SigMMD_naive_55121610277610
// MI455X (gfx1250) — compile-verified
//
#include <hip/hip_runtime.h>
#include <hip/hip_bf16.h>

// ---------------------------------------------------------------------------
// Signature-kernel MMD, fully fused on gfx1250 (MI455X).
//   X, Y: (64, 64, 128) f32.  Output: scalar f32.
// Per block (1 wave32): a 16x16 (a,b) tile of one Gram matrix.
//   for m in 0..62:                      (PDE row index)
//     load A-fragments afr[32] = dX tile row (diff folded into load)
//     for n in 0..62:                    (PDE col index, sequential scan)
//       inc_tile(16x16) = sum_{kc=0..31} WMMA_F32_16X16X4_F32(afr[kc], bfr)
//       PDE column step on 256 pairs (8 pairs/lane, prev row in LDS)
// WMMA f32 (not bf16): PDE compounds over 63 steps; keep full precision.
// ---------------------------------------------------------------------------

typedef __attribute__((ext_vector_type(2))) float v2f;
typedef __attribute__((ext_vector_type(8))) float v8f;

#define A_DIM 64
#define M_DIM 64
#define D_DIM 128
#define M1    63                 // diff length (time steps - 1)
#define TILE  16
#define NT    (A_DIM / TILE)     // 4 tiles per matrix side
#define BLOCKS_PER_GRAM (NT * NT)          // 16
#define N_BLOCKS (3 * BLOCKS_PER_GRAM)     // 48
#define NSLOT 256                // pairs per wave (16x16 tile)

__global__ void sig_zero_out(float* out) {
    if (threadIdx.x == 0) out[0] = 0.0f;
}

__global__ __launch_bounds__(32)
void sig_pde_wmma(const float* __restrict__ X, const float* __restrict__ Y,
                  float* __restrict__ out, float* __restrict__ ws, int use_ws)
{
    // Previous PDE row, n-major: Krow[n][slot]; 64*256*4 = 64KB LDS.
    // At fixed (r,n) lanes hit consecutive dwords -> conflict-free.
    __shared__ float Krow[M1 + 1][NSLOT];

    const int blk = blockIdx.x;
    const int g   = blk / BLOCKS_PER_GRAM;   // 0: K_XX, 1: K_YY, 2: K_XY
    const int t   = blk % BLOCKS_PER_GRAM;
    const int a0  = (t / NT) * TILE;
    const int b0  = (t % NT) * TILE;
    const float* P = (g == 1) ? Y : X;       // A-side paths
    const float* Q = (g == 0) ? X : Y;       // B-side paths
    const float w  = (g == 2) ? -2.0f : 1.0f;

    const int lane  = threadIdx.x;           // wave32
    const int row16 = lane & 15;             // M index (A) / N index (B)
    const int dhalf = (lane >> 4) * 2;       // K offset for upper lane half

    // PDE boundary: row i=0 is all ones (K[0][j] = 1); column j=0 stays 1.
    for (int idx = lane; idx < (M1 + 1) * NSLOT; idx += 32)
        (&Krow[0][0])[idx] = 1.0f;
    __syncthreads();

    // Per-lane base pointers. A frag element (M=row16, K=dhalf+v) at
    // d = 4*kc + dhalf + v  -> one float2 per chunk. Same shape for B.
    const float* Abase = P + ((size_t)(a0 + row16) * M_DIM) * D_DIM + dhalf;
    const float* Bbase = Q + ((size_t)(b0 + row16) * M_DIM) * D_DIM + dhalf;

    for (int m = 0; m < M1; ++m) {
        // A fragments for this PDE row: dX[a, m, :] (diff on the fly).
        const float* Am0 = Abase + m * D_DIM;
        v2f afr[32];
#pragma unroll
        for (int kc = 0; kc < 32; ++kc) {
            float2 x0 = *(const float2*)(Am0 + kc * 4);
            float2 x1 = *(const float2*)(Am0 + D_DIM + kc * 4);
            v2f a; a[0] = x1.x - x0.x; a[1] = x1.y - x0.y;
            afr[kc] = a;
        }
        if (m + 2 < M_DIM)                        // gfx1250 global_prefetch_b8
            __builtin_prefetch(Am0 + 2 * D_DIM, 0, 1);

        // Scan state per owned pair: k_left = K[i+1][j], k_diag = K[i][j].
        float kleft[8], kdiag[8];
#pragma unroll
        for (int r = 0; r < 8; ++r) { kleft[r] = 1.0f; kdiag[r] = 1.0f; }

        for (int n = 0; n < M1; ++n) {
            // inc tile for (m,n): 16x16 over (a,b), K-reduce over D=128.
            v8f acc = {};
            const float* Bn0 = Bbase + n * D_DIM;
#pragma unroll
            for (int kc = 0; kc < 32; ++kc) {
                float2 y0 = *(const float2*)(Bn0 + kc * 4);
                float2 y1 = *(const float2*)(Bn0 + D_DIM + kc * 4);
                v2f b; b[0] = y1.x - y0.x; b[1] = y1.y - y0.y;
                acc = __builtin_amdgcn_wmma_f32_16x16x4_f32(
                    /*neg_a=*/false, afr[kc], /*neg_b=*/false, b,
                    /*c_mod=*/(short)0, acc, /*reuse_a=*/false, /*reuse_b=*/false);
            }
            // PDE column step; slot (r,lane) <-> accumulator element layout.
#pragma unroll
            for (int r = 0; r < 8; ++r) {
                const int slot = r * 32 + lane;
                float kup  = Krow[n + 1][slot];                 // K[i][j+1]
                float knew = kleft[r] + kup + kdiag[r] * (acc[r] - 1.0f);
                Krow[n + 1][slot] = knew;                       // K[i+1][j+1]
                kdiag[r] = kup;
                kleft[r] = knew;
            }
        }
    }
    __syncthreads();

    // Tile result = sum over 256 pairs of K[-1][-1] = Krow[M1][slot].
    float part = 0.0f;
#pragma unroll
    for (int r = 0; r < 8; ++r) part += Krow[M1][r * 32 + lane];
    for (int off = 16; off > 0; off >>= 1)
        part += __shfl_down(part, off, 32);

    if (lane == 0) {
        if (use_ws) ws[blk] = part;     // deterministic path
        else atomicAdd(out, w * part * (1.0f / (A_DIM * A_DIM)));
    }
}

__global__ void sig_reduce(const float* __restrict__ ws, float* __restrict__ out) {
    if (threadIdx.x != 0) return;
    float total = 0.0f;
    for (int b = 0; b < N_BLOCKS; ++b) {
        float wgt = (b >= 2 * BLOCKS_PER_GRAM) ? -2.0f : 1.0f;
        total += wgt * ws[b];
    }
    out[0] = total * (1.0f / (A_DIM * A_DIM));
}

extern "C" void kernel_launch(void* const* d_in, const int* in_sizes, int n_in,
                              void* d_out, int out_size, void* d_ws, size_t ws_size,
                              hipStream_t stream) {
    const float* X = (const float*)d_in[0];
    const float* Y = (const float*)d_in[1];
    float* out = (float*)d_out;
    float* ws  = (float*)d_ws;
    const int use_ws = (ws != nullptr && ws_size >= N_BLOCKS * sizeof(float)) ? 1 : 0;

    sig_zero_out<<<1, 32, 0, stream>>>(out);
    sig_pde_wmma<<<N_BLOCKS, 32, 0, stream>>>(X, Y, out, ws, use_ws);
    if (use_ws)
        sig_reduce<<<1, 32, 0, stream>>>(ws, out);
}